// UnicycleIntegrator_77120432767634
// MI455X (gfx1250) — compile-verified
//
#include <hip/hip_runtime.h>
#include <cstdint>
#include <cstddef>

// Unicycle trajectory integrator for MI455X (gfx1250).
// Memory-bound: ~525 MB moved @ 23.3 TB/s => ~22.5 us floor.
//
// v2: wave-decoupled async pipeline. Each wave32 stages (via CDNA5
// global_load_async_to_lds_b128 / ASYNCcnt) exactly the 32 batch elements its
// own lanes integrate, double-buffered across time chunks. No workgroup
// barriers anywhere — the handshake is a per-wave s_wait_asynccnt.

#define TSTEPS   200
#define TC       8                   // timesteps per chunk
#define NCHUNK   (TSTEPS / TC)       // 25
#define BLOCK    256                 // 8 wave32s
#define NWAVE    (BLOCK / 32)
#define DT_F     0.1f

// LDS: [buf 0..1][wave 0..7][piece 0..3][lane-b 0..31] of float4 (v0,om0,v1,om1)
// per wave per buffer: 4*32*16 = 2 KB;  total = 2*8*2KB = 32 KB.
#define F4_PER_WAVEBUF (4 * 32)                  // 128 float4
#define F4_PER_BUF     (NWAVE * F4_PER_WAVEBUF)  // 1024 float4

__device__ __forceinline__ void async_b128_to_lds(uint32_t lds_off,
                                                  const float* __restrict__ base,
                                                  uint32_t g_off) {
  // GVS mode: 64-bit SGPR base + 32-bit per-lane byte offset.
  // VDST supplies the (workgroup-relative) LDS byte address.
  asm volatile("global_load_async_to_lds_b128 %0, %1, %2"
               :
               : "v"(lds_off), "v"(g_off), "s"(base)
               : "memory");
}

__global__ __launch_bounds__(BLOCK)
void UnicycleIntegrator_kernel(const float* __restrict__ v_omega,   // (B, T, 2)
                               const float* __restrict__ init_state,// (B, 3)
                               float* __restrict__ out,             // (B, T, 3)
                               int B) {
  __shared__ float4 smem[2 * F4_PER_BUF];

  const int tid  = (int)threadIdx.x;
  const int wave = tid >> 5;          // 0..7
  const int lane = tid & 31;          // 0..31
  const int b    = (int)blockIdx.x * BLOCK + tid;   // this lane's trajectory

  // Staging role within the wave: piece = which 16B (2 timesteps) fragment,
  // bb0 = which of the wave's 32 batch elements (advances by 8 per issue k).
  const int piece = lane & 3;         // 0..3
  const int bb0   = lane >> 2;        // 0..7
  const int wave_b0 = (int)blockIdx.x * BLOCK + (wave << 5);  // first b of wave

  // Workgroup-relative LDS byte offset (low 32 bits of the flat pointer).
  const uint32_t lds_base = (uint32_t)(uintptr_t)(void*)&smem[0];
  const uint32_t wave_lds = lds_base + (uint32_t)(wave * F4_PER_WAVEBUF * 16);

  // --- load initial state (one-time, tiny) -------------------------------
  float x, y, th;
  {
    int bc = b < B ? b : (B - 1);
    size_t ib = (size_t)bc * 3;
    x  = init_state[ib + 0];
    y  = init_state[ib + 1];
    th = init_state[ib + 2];
  }

  // --- per-wave issue: stage chunk c (2 KB for this wave) into buffer buf
  auto issue_chunk = [&](int c, int buf) {
#pragma unroll
    for (int k = 0; k < 4; ++k) {
      int bb = bb0 + (k << 3);                 // 0..31 within the wave
      int bg = wave_b0 + bb;                   // global batch element
      if (bg >= B) bg = B - 1;                 // clamp (B % 256 == 0 anyway)
      uint32_t goff = (uint32_t)bg * (uint32_t)(TSTEPS * 8)
                    + (uint32_t)c * (uint32_t)(TC * 8)
                    + (uint32_t)piece * 16u;
      uint32_t loff = wave_lds
                    + (uint32_t)buf * (uint32_t)(F4_PER_BUF * 16)
                    + (uint32_t)piece * (uint32_t)(32 * 16)
                    + (uint32_t)bb * 16u;
      async_b128_to_lds(loff, v_omega, goff);
    }
  };

  // --- prime the wave-private pipeline -----------------------------------
  issue_chunk(0, 0);

  const float dt = DT_F;

  for (int c = 0; c < NCHUNK; ++c) {
    const int cur = c & 1;

    if (c + 1 < NCHUNK) {
      issue_chunk(c + 1, cur ^ 1);
      // 4 newest (next chunk) may stay outstanding; current chunk is landed.
      asm volatile("s_wait_asynccnt 0x4" ::: "memory");
    } else {
      asm volatile("s_wait_asynccnt 0x0" ::: "memory");
    }
    // No barrier: this wave only reads LDS it staged itself.

    // --- consume 8 timesteps from LDS, integrate, buffer 24 output floats
    const float4* sbuf = &smem[cur * F4_PER_BUF + wave * F4_PER_WAVEBUF];
    float o[TC * 3];
#pragma unroll
    for (int p = 0; p < 4; ++p) {
      float4 q = sbuf[p * 32 + lane];   // (v0, om0, v1, om1)

      // step 2p
      th += q.y * dt;
      float sn, cs;
      __sincosf(th, &sn, &cs);
      float vdt = q.x * dt;
      x += vdt * cs;
      y += vdt * sn;
      o[p * 6 + 0] = x; o[p * 6 + 1] = y; o[p * 6 + 2] = th;

      // step 2p+1
      th += q.w * dt;
      __sincosf(th, &sn, &cs);
      vdt = q.z * dt;
      x += vdt * cs;
      y += vdt * sn;
      o[p * 6 + 3] = x; o[p * 6 + 4] = y; o[p * 6 + 5] = th;
    }

    // --- write chunk: 6 aligned 16B stores, lane-contiguous 96B slice.
    // L2 (192 MB) merges the per-lane slices into full-line HBM writes.
    if (b < B) {
      float4* op = (float4*)(out + (size_t)b * (TSTEPS * 3) + (size_t)c * (TC * 3));
#pragma unroll
      for (int i = 0; i < 6; ++i) {
        op[i] = make_float4(o[4 * i + 0], o[4 * i + 1], o[4 * i + 2], o[4 * i + 3]);
      }
    }
    // WAR on buffer `cur` is safe: the ds_load results above were consumed by
    // VALU (DScnt drained) before next iteration's async issue in program order.
  }
}

extern "C" void kernel_launch(void* const* d_in, const int* in_sizes, int n_in,
                              void* d_out, int out_size, void* d_ws, size_t ws_size,
                              hipStream_t stream) {
  (void)n_in; (void)d_ws; (void)ws_size; (void)out_size;
  const float* v_omega    = (const float*)d_in[0];   // (B, T, 2) fp32
  const float* init_state = (const float*)d_in[1];   // (B, 3)    fp32
  float*       out        = (float*)d_out;           // (B, T, 3) fp32

  const int B = in_sizes[1] / 3;                     // 131072
  const int grid = (B + BLOCK - 1) / BLOCK;          // 512 blocks of 8 wave32s

  hipLaunchKernelGGL(UnicycleIntegrator_kernel, dim3(grid), dim3(BLOCK), 0, stream,
                     v_omega, init_state, out, B);
}